// CustomLoss_71837622993168
// MI455X (gfx1250) — compile-verified
//
#include <hip/hip_runtime.h>
#include <hip/hip_bf16.h>

typedef __attribute__((ext_vector_type(2))) float v2f;
typedef __attribute__((ext_vector_type(8))) float v8f;

#define OUTD 58
#define SSIM_C1 1.0e-4f
#define SSIM_C2 9.0e-4f
#define COV_NORM (49.0f/48.0f)

// Banded box-filter matrix V (64x64, rows >= 58 are zero):
// V[r][k] = 1 if r<58 and r <= k <= r+6
__device__ __forceinline__ float vband(int r, int k) {
  return (r < OUTD && k >= r && k <= r + 6) ? 1.0f : 0.0f;
}

// One block per (b, c, pair). 128 threads = 4 wave32s.
// Computes sum over the 58x58 SSIM map for this (b,pair,c) into part[].
__global__ __launch_bounds__(128) void ssim_pair_kernel(
    const float* __restrict__ features, float* __restrict__ part)
{
  __shared__ __align__(16) float Xi[64 * 64];
  __shared__ __align__(16) float Xj[64 * 64];
  __shared__ __align__(16) float Tq[5][16 * 64];   // V*M strips (16 rows x 64 cols)
  __shared__ float red[128];

  const int tid  = threadIdx.x;
  const int lane = tid & 31;
  const int wave = tid >> 5;      // 0..3 -> output column tile
  const int half = lane >> 4;     // 0/1  -> K-group within fragment
  const int lm   = lane & 15;     // M (A rows) or N (B,C/D cols)

  const int p = blockIdx.x % 28;
  const int c = (blockIdx.x / 28) % 16;
  const int b = blockIdx.x / (28 * 16);

  // np.triu_indices(8, k=1) row-major order -> (fi, fj)
  int fi = 0, rem = p, cnt = 7;
  while (rem >= cnt) { rem -= cnt; ++fi; --cnt; }
  const int fj = fi + 1 + rem;

  // Load the two 64x64 f32 maps into LDS (vectorized b128).
  const float4* si = (const float4*)(features + (((size_t)b * 8 + fi) * 16 + c) * 4096);
  const float4* sj = (const float4*)(features + (((size_t)b * 8 + fj) * 16 + c) * 4096);
  float4* di = (float4*)Xi;
  float4* dj = (float4*)Xj;
  for (int t = tid; t < 1024; t += 128) { di[t] = si[t]; dj[t] = sj[t]; }

  const int col = wave * 16 + lm;   // global output/input column this lane touches
  float acc = 0.0f;

  for (int tm = 0; tm < 4; ++tm) {
    __syncthreads();   // load done / previous stage-2 reads of Tq done

    // ---- Stage 1: T_q[tm-strip] = V(rows tm*16..+15) * M_q  (5 moment maps) ----
    v8f tacc[5];
    #pragma unroll
    for (int q = 0; q < 5; ++q) tacc[q] = (v8f){0,0,0,0,0,0,0,0};

    const int row = tm * 16 + lm;   // A-matrix row (V row)
    #pragma unroll
    for (int k0 = 0; k0 < 64; k0 += 4) {
      const int kk = k0 + 2 * half;
      v2f a; a.x = vband(row, kk); a.y = vband(row, kk + 1);

      const float xi0 = Xi[kk * 64 + col];
      const float xi1 = Xi[(kk + 1) * 64 + col];
      const float xj0 = Xj[kk * 64 + col];
      const float xj1 = Xj[(kk + 1) * 64 + col];
      const float m0[5] = { xi0, xj0, xi0 * xi0, xj0 * xj0, xi0 * xj0 };
      const float m1[5] = { xi1, xj1, xi1 * xi1, xj1 * xj1, xi1 * xj1 };

      #pragma unroll
      for (int q = 0; q < 5; ++q) {
        v2f bb; bb.x = m0[q]; bb.y = m1[q];
        tacc[q] = __builtin_amdgcn_wmma_f32_16x16x4_f32(
            false, a, false, bb, (short)0, tacc[q], false, false);
      }
    }
    #pragma unroll
    for (int r = 0; r < 8; ++r) {
      const int m = r + half * 8;        // C/D row layout
      #pragma unroll
      for (int q = 0; q < 5; ++q) Tq[q][m * 64 + col] = tacc[q][r];
    }
    __syncthreads();

    // ---- Stage 2: S_q(tile tm, tn=wave) = T_q * V^T ----
    v8f sacc[5];
    #pragma unroll
    for (int q = 0; q < 5; ++q) sacc[q] = (v8f){0,0,0,0,0,0,0,0};

    #pragma unroll
    for (int k0 = 0; k0 < 64; k0 += 4) {
      const int kk = k0 + 2 * half;
      v2f bb; bb.x = vband(col, kk); bb.y = vband(col, kk + 1);  // V^T[k][n] = V[n][k]
      #pragma unroll
      for (int q = 0; q < 5; ++q) {
        v2f aa; aa.x = Tq[q][lm * 64 + kk]; aa.y = Tq[q][lm * 64 + kk + 1];
        sacc[q] = __builtin_amdgcn_wmma_f32_16x16x4_f32(
            false, aa, false, bb, (short)0, sacc[q], false, false);
      }
    }

    // ---- SSIM on this 16x16 tile (mask to valid 58x58 region) ----
    #pragma unroll
    for (int r = 0; r < 8; ++r) {
      const int m = tm * 16 + r + half * 8;
      if (m < OUTD && col < OUTD) {
        const float inv = 1.0f / 49.0f;
        const float ux  = sacc[0][r] * inv;
        const float uy  = sacc[1][r] * inv;
        const float uxx = sacc[2][r] * inv;
        const float uyy = sacc[3][r] * inv;
        const float uxy = sacc[4][r] * inv;
        const float vx  = COV_NORM * (uxx - ux * ux);
        const float vy  = COV_NORM * (uyy - uy * uy);
        const float vxy = COV_NORM * (uxy - ux * uy);
        const float S = ((2.0f * ux * uy + SSIM_C1) * (2.0f * vxy + SSIM_C2)) /
                        ((ux * ux + uy * uy + SSIM_C1) * (vx + vy + SSIM_C2));
        acc += S;
      }
    }
  }

  // Block reduction -> one partial per (b, pair, c); deterministic (no atomics).
  __syncthreads();
  red[tid] = acc;
  __syncthreads();
  for (int s = 64; s > 0; s >>= 1) {
    if (tid < s) red[tid] += red[tid + s];
    __syncthreads();
  }
  if (tid == 0) part[((size_t)b * 28 + p) * 16 + c] = red[0];
}

// One wave: combine SSIM partials + weighted CE into the scalar loss.
__global__ __launch_bounds__(32) void finalize_kernel(
    const float* __restrict__ pred, const int* __restrict__ labels,
    const float* __restrict__ part, float* __restrict__ out)
{
  __shared__ float s_wnll[32], s_w[32], s_bce[32];
  const int b = threadIdx.x;

  // mean SSIM -> sim -> avg over 28 pairs
  float avg = 0.0f;
  for (int p = 0; p < 28; ++p) {
    float s = 0.0f;
    for (int c = 0; c < 16; ++c) s += part[((size_t)b * 28 + p) * 16 + c];
    const float sp  = s * (1.0f / (16.0f * 58.0f * 58.0f));
    const float sim = fminf(fmaxf(sp + 0.5f, 0.0f), 1.0f);
    avg += sim;
  }
  avg *= (1.0f / 28.0f);

  const int   l  = labels[b];
  const float p0 = pred[2 * b], p1 = pred[2 * b + 1];
  const float mx  = fmaxf(p0, p1);
  const float lse = mx + logf(expf(p0 - mx) + expf(p1 - mx));
  const float pl  = (l == 0) ? p0 : p1;
  const float w   = (l == 0) ? 10.0f : 1.0f;
  s_wnll[b] = w * (lse - pl);
  s_w[b]    = w;
  s_bce[b]  = (l == 0) ? -fmaxf(logf(avg), -100.0f)
                       : -fmaxf(logf(1.0f - avg), -100.0f);
  __syncthreads();
  for (int s = 16; s > 0; s >>= 1) {
    if (b < s) { s_wnll[b] += s_wnll[b + s]; s_w[b] += s_w[b + s]; s_bce[b] += s_bce[b + s]; }
    __syncthreads();
  }
  if (b == 0) {
    const float cce = s_wnll[0] / s_w[0];
    const float inc = s_bce[0] * (1.0f / 32.0f);
    out[0] = cce + 4.0f * inc;
  }
}

extern "C" void kernel_launch(void* const* d_in, const int* in_sizes, int n_in,
                              void* d_out, int out_size, void* d_ws, size_t ws_size,
                              hipStream_t stream) {
  const float* predictions = (const float*)d_in[0];
  const float* features    = (const float*)d_in[1];
  const int*   labels      = (const int*)d_in[2];
  float* out  = (float*)d_out;
  float* part = (float*)d_ws;   // 32*28*16 floats = 57,344 B of scratch

  ssim_pair_kernel<<<32 * 16 * 28, 128, 0, stream>>>(features, part);
  finalize_kernel<<<1, 32, 0, stream>>>(predictions, labels, part, out);
}